// ActivityAugmentation_86904368267728
// MI455X (gfx1250) — compile-verified
//
#include <hip/hip_runtime.h>
#include <stdint.h>

// B=512, T=2048, C=64, f32 everywhere.
typedef float v4f __attribute__((ext_vector_type(4)));

#define B_ 512
#define T_ 2048
#define C_ 64
#define TSHIFT 11     // T = 2^11
#define TMASK  2047

// ---------------------------------------------------------------------------
// Kernel 1: precompute time-warp gather indices i0[T] and fractions frac[T].
// One block of 256 threads, 8 elements/thread, Hillis-Steele scan in LDS.
// ---------------------------------------------------------------------------
__global__ __launch_bounds__(256) void warp_precompute_kernel(
    const float* __restrict__ warp_noise,
    int* __restrict__ i0_out,
    float* __restrict__ frac_out)
{
  __shared__ float partial[256];
  __shared__ float shw0;
  const int tid = threadIdx.x;
  const float k = 0.2f / 2048.0f;   // TW_SIGMA / T

  float vals[8];
  float s = 0.0f;
#pragma unroll
  for (int j = 0; j < 8; ++j) {
    vals[j] = warp_noise[tid * 8 + j] * k;
    s += vals[j];
  }
  partial[tid] = s;
  __syncthreads();

  // inclusive scan over the 256 per-thread sums
#pragma unroll
  for (int off = 1; off < 256; off <<= 1) {
    const float add = (tid >= off) ? partial[tid - off] : 0.0f;
    __syncthreads();
    partial[tid] += add;
    __syncthreads();
  }

  if (tid == 0) shw0 = vals[0];   // warp[0] = first cumsum element
  __syncthreads();

  const float w0    = shw0;
  const float total = partial[255];           // warp[T-1] before shift
  const float denom = (total - w0) + 1e-8f;   // reference: (warp[-1] after -w0) + 1e-8

  float run = (tid == 0) ? 0.0f : partial[tid - 1];   // exclusive prefix
#pragma unroll
  for (int j = 0; j < 8; ++j) {
    const int t = tid * 8 + j;
    run += vals[j];                            // inclusive cumsum at t
    const float wn = (run - w0) / denom;       // normalized warp
    float tw = (float)t / 2047.0f + wn * 0.2f; // t_orig + warp*0.2
    tw = fminf(fmaxf(tw, 0.0f), 1.0f);
    const float pos = tw * 2047.0f;
    int i = (int)floorf(pos);
    i = i < 0 ? 0 : (i > 2046 ? 2046 : i);
    i0_out[t]   = i;
    frac_out[t] = pos - (float)i;
  }
}

// ---------------------------------------------------------------------------
// Kernel 2: fused jitter + scale + time-warp lerp + rotate(ch0,ch1) + chmask.
// 16 lanes x float4 per output row; 16 rows per 256-thread block.
// Inputs staged global -> LDS via CDNA5 async copies (ASYNCcnt), output
// written with non-temporal 128-bit stores.
// ---------------------------------------------------------------------------
__global__ __launch_bounds__(256) void aug_main_kernel(
    const float* __restrict__ x, const float* __restrict__ noise,
    const float* __restrict__ scale_u, const float* __restrict__ angle_u,
    const float* __restrict__ chmask_u,
    const int* __restrict__ i0_arr, const float* __restrict__ frac_arr,
    float* __restrict__ out)
{
  __shared__ __attribute__((aligned(16))) float stage[256 * 16];  // 16 KB

  const int tid  = threadIdx.x;
  const int lane = tid & 15;        // which float4 chunk of the 64 channels
  const int row  = tid >> 4;        // 0..15
  const unsigned gr = blockIdx.x * 16u + (unsigned)row;   // global output row
  const unsigned b  = gr >> TSHIFT;
  const unsigned t  = gr & TMASK;

  const int   i = i0_arr[t];
  const float f = frac_arr[t];

  const unsigned ch     = (unsigned)lane * 4u;
  const unsigned base_f = (b * (unsigned)T_ + (unsigned)i) * (unsigned)C_ + ch;
  const unsigned byte0  = base_f * 4u;                 // row i0
  const unsigned byte1  = byte0 + (unsigned)(C_ * 4);  // row i0+1

  // LDS byte address of this thread's private 64B staging slot
  const unsigned lbase = (unsigned)(uintptr_t)&stage[tid * 16];

  // Async global->LDS copies (tracked by ASYNCcnt). GVS mode: SGPR base +
  // per-lane u32 byte offset in VADDR; VDST VGPR carries the LDS address.
  asm volatile("global_load_async_to_lds_b128 %0, %1, %2"
               :: "v"(lbase),       "v"(byte0), "s"(x)     : "memory");
  asm volatile("global_load_async_to_lds_b128 %0, %1, %2"
               :: "v"(lbase + 16u), "v"(byte1), "s"(x)     : "memory");
  asm volatile("global_load_async_to_lds_b128 %0, %1, %2"
               :: "v"(lbase + 32u), "v"(byte0), "s"(noise) : "memory");
  asm volatile("global_load_async_to_lds_b128 %0, %1, %2"
               :: "v"(lbase + 48u), "v"(byte1), "s"(noise) : "memory");

  // Overlap scalar/VALU work with the async copies.
  const float scale = 0.9f + scale_u[0] * 0.2f;
  const v4f   m     = *(const v4f*)(chmask_u + ch);
  float ca = 1.0f, sa = 0.0f;
  if (lane == 0) {
    const float a = angle_u[b] * 6.28318530717958647692f
                    - 3.14159265358979323846f;
    sincosf(a, &sa, &ca);
  }

  asm volatile("s_wait_asynccnt 0x0" ::: "memory");

  const v4f x0 = *(const v4f*)&stage[tid * 16 + 0];
  const v4f x1 = *(const v4f*)&stage[tid * 16 + 4];
  const v4f n0 = *(const v4f*)&stage[tid * 16 + 8];
  const v4f n1 = *(const v4f*)&stage[tid * 16 + 12];

  const v4f j0 = (x0 + n0 * 0.01f) * scale;
  const v4f j1 = (x1 + n1 * 0.01f) * scale;
  v4f v = j0 * (1.0f - f) + j1 * f;

  if (lane == 0) {  // rotate channels 0,1 by per-batch angle
    const float r0 = ca * v.x - sa * v.y;
    const float r1 = sa * v.x + ca * v.y;
    v.x = r0;
    v.y = r1;
  }

  v4f mm;
  mm.x = (m.x > 0.1f) ? 1.0f : 0.0f;
  mm.y = (m.y > 0.1f) ? 1.0f : 0.0f;
  mm.z = (m.z > 0.1f) ? 1.0f : 0.0f;
  mm.w = (m.w > 0.1f) ? 1.0f : 0.0f;
  v *= mm;

  const unsigned ooff = gr * (unsigned)C_ + ch;
  __builtin_nontemporal_store(v, (v4f*)(out + ooff));  // write-once output
}

// ---------------------------------------------------------------------------
extern "C" void kernel_launch(void* const* d_in, const int* in_sizes, int n_in,
                              void* d_out, int out_size, void* d_ws, size_t ws_size,
                              hipStream_t stream) {
  (void)in_sizes; (void)n_in; (void)out_size; (void)ws_size;
  const float* x       = (const float*)d_in[0];
  const float* noise   = (const float*)d_in[1];
  const float* scale_u = (const float*)d_in[2];
  const float* warp_n  = (const float*)d_in[3];
  const float* angle_u = (const float*)d_in[4];
  const float* chmask  = (const float*)d_in[5];
  float* out = (float*)d_out;

  int*   i0_arr   = (int*)d_ws;                                  // 2048 ints
  float* frac_arr = (float*)((char*)d_ws + T_ * sizeof(int));    // 2048 floats

  warp_precompute_kernel<<<1, 256, 0, stream>>>(warp_n, i0_arr, frac_arr);
  aug_main_kernel<<<(B_ * T_) / 16, 256, 0, stream>>>(
      x, noise, scale_u, angle_u, chmask, i0_arr, frac_arr, out);
}